// GraphClassificationModelCrossEntropy_2284922601619
// MI455X (gfx1250) — compile-verified
//
#include <hip/hip_runtime.h>

// ---------------------------------------------------------------------------
// 2-layer GCN, reordered as GEMM-first:
//   Y1   = (out_s ⊙ X) @ W1                    (WMMA f32 16x16x4)
//   agg1 = scatter_add(Y1[src] -> dst)         (global_atomic_add_f32, 64 f/edge)
//   A2   = out_s ⊙ relu(in_s ⊙ agg1 + b1)      (fused into GEMM2 A-load)
//   Z2   = A2 @ W2                             (WMMA f32 16x16x4)
//   agg2 = scatter_add(Z2[src] -> dst)         (16 f/edge)
//   out  = in_s ⊙ agg2 + b2
// All intermediates (<33 MB) live in d_ws and stay resident in the 192 MB L2.
// ---------------------------------------------------------------------------

typedef float v2f __attribute__((ext_vector_type(2)));
typedef float v8f __attribute__((ext_vector_type(8)));

#define IN_DIM 128
#define HID    64
#define OUT_DIM 16

// ---- degrees: one f32 atomic per endpoint (3.2M atomics, L2 resident) -----
__global__ void degree_kernel(const int* __restrict__ src,
                              const int* __restrict__ dst,
                              float* __restrict__ outdeg,
                              float* __restrict__ indeg, int nE) {
  int e = blockIdx.x * blockDim.x + threadIdx.x;
  if (e >= nE) return;
  unsafeAtomicAdd(&outdeg[src[e]], 1.0f);
  unsafeAtomicAdd(&indeg[dst[e]], 1.0f);
}

// ---- deg -> clip(.,1)^-1/2 in place ---------------------------------------
__global__ void invsqrt_kernel(float* __restrict__ d, int n) {
  int i = blockIdx.x * blockDim.x + threadIdx.x;
  if (i >= n) return;
  float v = d[i];
  v = v < 1.0f ? 1.0f : v;
  d[i] = rsqrtf(v);
}

// ---- GEMM1: Y[N x 64] = (out_s ⊙ X[N x 128]) @ W1[128 x 64] ---------------
// One wave per 16x16 tile. Block = 256 thr = 8 waves = 2 Mtiles x 4 Ntiles.
__global__ void gemm1_kernel(const float* __restrict__ X,
                             const float* __restrict__ W1,
                             const float* __restrict__ out_s,
                             float* __restrict__ Y, int numMt) {
  const int wave = threadIdx.x >> 5;
  const int lane = threadIdx.x & 31;
  const int mt = blockIdx.x * 2 + (wave >> 2);   // M tile
  const int nt = wave & 3;                        // N tile (HID/16 = 4)
  if (mt >= numMt) return;                        // wave-uniform: EXEC stays all-1

  const int row  = mt * 16 + (lane & 15);         // A: M = lane&15
  const int koff = (lane >> 4) * 2;               // A/B: K = 2*(lane>>4)+{0,1}
  const float s = out_s[row];
  const float* __restrict__ arow = X + (size_t)row * IN_DIM;
  const float* __restrict__ bcol = W1 + nt * 16 + (lane & 15);

  v8f c = {};
  for (int k0 = 0; k0 < IN_DIM; k0 += 4) {
    v2f a, b;
    a.x = arow[k0 + koff]     * s;
    a.y = arow[k0 + koff + 1] * s;
    b.x = bcol[(size_t)(k0 + koff)     * HID];
    b.y = bcol[(size_t)(k0 + koff + 1) * HID];
    c = __builtin_amdgcn_wmma_f32_16x16x4_f32(false, a, false, b,
                                              (short)0, c, false, false);
  }
  // C/D layout: VGPR i -> M = i + 8*(lane>=16), N = lane&15
  const int mbase = mt * 16 + (lane >> 4) * 8;
  const int col   = nt * 16 + (lane & 15);
#pragma unroll
  for (int i = 0; i < 8; ++i)
    Y[(size_t)(mbase + i) * HID + col] = c[i];
}

// ---- scatter 64 floats/edge: one wave per edge, float2 per lane -----------
__global__ void scatter64_kernel(const float* __restrict__ Y,
                                 const int* __restrict__ src,
                                 const int* __restrict__ dst,
                                 float* __restrict__ agg, int nE) {
  int t = blockIdx.x * blockDim.x + threadIdx.x;
  int e = t >> 5, lane = t & 31;
  if (e >= nE) return;
  int s = src[e], d = dst[e];
  float2 v = ((const float2*)(Y + (size_t)s * HID))[lane];
  float* p = agg + (size_t)d * HID + lane * 2;
  unsafeAtomicAdd(p,     v.x);
  unsafeAtomicAdd(p + 1, v.y);
}

// ---- GEMM2: Z[N x 16] = (out_s ⊙ relu(in_s ⊙ agg1 + b1)) @ W2[64 x 16] ----
__global__ void gemm2_kernel(const float* __restrict__ agg1,
                             const float* __restrict__ W2,
                             const float* __restrict__ in_s,
                             const float* __restrict__ out_s,
                             const float* __restrict__ b1,
                             float* __restrict__ Z, int numMt) {
  const int wave = threadIdx.x >> 5;
  const int lane = threadIdx.x & 31;
  const int mt = blockIdx.x * 8 + wave;
  if (mt >= numMt) return;                        // wave-uniform

  const int row  = mt * 16 + (lane & 15);
  const int koff = (lane >> 4) * 2;
  const float si = in_s[row];
  const float so = out_s[row];
  const float* __restrict__ arow = agg1 + (size_t)row * HID;
  const float* __restrict__ bcol = W2 + (lane & 15);

  v8f c = {};
  for (int k0 = 0; k0 < HID; k0 += 4) {
    int k = k0 + koff;
    float a0 = fmaxf(arow[k]     * si + b1[k],     0.0f) * so;
    float a1 = fmaxf(arow[k + 1] * si + b1[k + 1], 0.0f) * so;
    v2f a = {a0, a1};
    v2f b = {bcol[(size_t)k * OUT_DIM], bcol[(size_t)(k + 1) * OUT_DIM]};
    c = __builtin_amdgcn_wmma_f32_16x16x4_f32(false, a, false, b,
                                              (short)0, c, false, false);
  }
  const int mbase = mt * 16 + (lane >> 4) * 8;
  const int col   = lane & 15;
#pragma unroll
  for (int i = 0; i < 8; ++i)
    Z[(size_t)(mbase + i) * OUT_DIM + col] = c[i];
}

// ---- scatter 16 floats/edge: 16 lanes per edge ----------------------------
__global__ void scatter16_kernel(const float* __restrict__ Z,
                                 const int* __restrict__ src,
                                 const int* __restrict__ dst,
                                 float* __restrict__ agg, int nE) {
  int t = blockIdx.x * blockDim.x + threadIdx.x;
  int e = t >> 4, cidx = t & 15;
  if (e >= nE) return;
  int s = src[e], d = dst[e];
  unsafeAtomicAdd(agg + (size_t)d * OUT_DIM + cidx,
                  Z[(size_t)s * OUT_DIM + cidx]);
}

// ---- out = in_s ⊙ agg2 + b2 ----------------------------------------------
__global__ void finalize_kernel(const float* __restrict__ agg2,
                                const float* __restrict__ in_s,
                                const float* __restrict__ b2,
                                float* __restrict__ out, int total) {
  int i = blockIdx.x * blockDim.x + threadIdx.x;
  if (i >= total) return;
  int node = i >> 4, cidx = i & 15;
  out[i] = agg2[i] * in_s[node] + b2[cidx];
}

extern "C" void kernel_launch(void* const* d_in, const int* in_sizes, int n_in,
                              void* d_out, int out_size, void* d_ws, size_t ws_size,
                              hipStream_t stream) {
  const float* X   = (const float*)d_in[0];
  const int*   src = (const int*)  d_in[1];
  const int*   dst = (const int*)  d_in[2];
  const float* W1  = (const float*)d_in[3];
  const float* b1  = (const float*)d_in[4];
  const float* W2  = (const float*)d_in[5];
  const float* b2  = (const float*)d_in[6];
  float*       out = (float*)d_out;

  const int N = in_sizes[0] / IN_DIM;   // 50000
  const int E = in_sizes[1];            // 1600000
  const int numMt = (N + 15) / 16;      // 3125 (exact: N % 16 == 0)

  // workspace layout (floats)
  float* ws     = (float*)d_ws;
  float* outdeg = ws;                    // N   (becomes out_s in place)
  float* indeg  = ws + N;                // N   (becomes in_s in place)
  float* Y1     = ws + 2 * (size_t)N;    // N*64
  float* agg1   = Y1 + (size_t)N * HID;  // N*64
  float* Z2     = agg1 + (size_t)N * HID;// N*16
  float* agg2   = Z2 + (size_t)N * OUT_DIM; // N*16

  // zero the accumulation targets every call (graph-capturable memset nodes)
  hipMemsetAsync(outdeg, 0, (size_t)2 * N * sizeof(float), stream);
  hipMemsetAsync(agg1,   0, (size_t)N * HID * sizeof(float), stream);
  hipMemsetAsync(agg2,   0, (size_t)N * OUT_DIM * sizeof(float), stream);

  const int T = 256;
  degree_kernel<<<(E + T - 1) / T, T, 0, stream>>>(src, dst, outdeg, indeg, E);
  invsqrt_kernel<<<(2 * N + T - 1) / T, T, 0, stream>>>(outdeg, 2 * N);

  gemm1_kernel<<<(numMt + 1) / 2, T, 0, stream>>>(X, W1, outdeg, Y1, numMt);
  scatter64_kernel<<<((size_t)E * 32 + T - 1) / T, T, 0, stream>>>(Y1, src, dst, agg1, E);

  gemm2_kernel<<<(numMt + 7) / 8, T, 0, stream>>>(agg1, W2, indeg, outdeg, b1, Z2, numMt);
  scatter16_kernel<<<((size_t)E * 16 + T - 1) / T, T, 0, stream>>>(Z2, src, dst, agg2, E);

  finalize_kernel<<<((size_t)N * OUT_DIM + T - 1) / T, T, 0, stream>>>(agg2, indeg, b2, out, N * OUT_DIM);
}